// SlidingAttention_10230612099674
// MI455X (gfx1250) — compile-verified
//
#include <hip/hip_runtime.h>

typedef float v2f __attribute__((ext_vector_type(2)));
typedef float v8f __attribute__((ext_vector_type(8)));

#define SEQ 4096
#define WIN 256
#define NC  513          // 2*WIN+1 band width
#define NT  33           // (2*WIN+16)/16 key tiles covering the band span
#define LDS_STRIDE 530   // 528 cols + skew pad; 8*530 % 64 == 16 -> bank-disjoint halves

__global__ __launch_bounds__(128)
void sliding_qk_softmax_kernel(const float* __restrict__ x, float* __restrict__ out) {
    __shared__ float sc[16 * LDS_STRIDE];

    const int i0   = blockIdx.x * 16;      // query tile start within sequence
    const int p    = blockIdx.y;           // sequence index (0..31)
    const int lane = threadIdx.x & 31;
    const int wv   = threadIdx.x >> 5;     // wave id 0..3
    const int n    = lane & 15;
    const int hi   = lane >> 4;            // half-wave selector

    const float* base = x + (size_t)p * SEQ * 64;

    // ---- A operand (Q tile): 16 steps of 16x16x4 across D=64.
    // A layout (ISA 16x4 f32): lane m<16 holds {K=4k, 4k+1}; lane m>=16 holds {K=4k+2, 4k+3}.
    v2f aQ[16];
    {
        const float* qp = base + (i0 + n) * 64 + 2 * hi;
        #pragma unroll
        for (int k = 0; k < 16; ++k)
            aQ[k] = *(const v2f*)(qp + 4 * k);
    }

    // ---- score tiles across the 528-wide band span, skew-stored into LDS
    for (int t = wv; t < NT; t += 4) {
        int j  = i0 - WIN + t * 16 + n;                 // key row for this lane's column
        int jc = j < 0 ? 0 : (j > SEQ - 1 ? SEQ - 1 : j); // clamp; masked at softmax
        const float* kp = base + jc * 64 + 2 * hi;       // B layout mirrors A layout

        // prefetch this wave's next key tile (t+4 -> +64 rows)
        int jn = jc + 64; if (jn > SEQ - 1) jn = SEQ - 1;
        __builtin_prefetch(base + jn * 64, 0, 0);

        v8f acc = {0.f, 0.f, 0.f, 0.f, 0.f, 0.f, 0.f, 0.f};
        #pragma unroll
        for (int k = 0; k < 16; ++k) {
            v2f bv = *(const v2f*)(kp + 4 * k);
            acc = __builtin_amdgcn_wmma_f32_16x16x4_f32(
                false, aQ[k], false, bv, (short)0, acc, false, false);
        }

        // C/D layout: VGPR d -> row d (lanes 0-15) / row d+8 (lanes 16-31), col = lane&15
        #pragma unroll
        for (int d = 0; d < 8; ++d)
            sc[(d + 8 * hi) * LDS_STRIDE + t * 16 + n] = acc[d];
    }

    __syncthreads();

    // ---- per-row fp32 softmax over the 513-wide band (skewed read: col = c + r)
    for (int r = wv; r < 16; r += 4) {
        const int i = i0 + r;
        const float* srow = sc + r * LDS_STRIDE + r;    // band element c at srow[c]

        float m = -INFINITY, sum = 0.f;
        for (int c = lane; c < NC; c += 32) {
            int j = i + c - WIN;
            if ((unsigned)j < (unsigned)SEQ) {
                float v  = srow[c];
                float nm = fmaxf(m, v);
                sum = sum * __expf(m - nm) + __expf(v - nm);
                m = nm;
            }
        }
        #pragma unroll
        for (int off = 16; off > 0; off >>= 1) {
            float mo = __shfl_xor(m, off, 32);
            float so = __shfl_xor(sum, off, 32);
            float nm = fmaxf(m, mo);
            sum = sum * __expf(m - nm) + so * __expf(mo - nm);
            m = nm;
        }
        float inv = 1.0f / sum;

        const int b = p >> 4, h = p & 15;
        float* orow = out + ((size_t)(b * SEQ + i) * 16 + h) * NC;
        for (int c = lane; c < NC; c += 32) {
            int j = i + c - WIN;
            orow[c] = ((unsigned)j < (unsigned)SEQ) ? __expf(srow[c] - m) * inv : 0.f;
        }
    }
}

extern "C" void kernel_launch(void* const* d_in, const int* in_sizes, int n_in,
                              void* d_out, int out_size, void* d_ws, size_t ws_size,
                              hipStream_t stream) {
    const float* x = (const float*)d_in[0];
    float* out = (float*)d_out;
    // number of flattened (B*H) sequences: total elems / (SEQ * D)
    int nP = in_sizes[0] / (SEQ * 64);     // = 32 for (2,16,4096,64)
    dim3 grid(SEQ / 16, nP);
    sliding_qk_softmax_kernel<<<grid, 128, 0, stream>>>(x, out);
}